// TotalLoss_42030549958920
// MI455X (gfx1250) — compile-verified
//
#include <hip/hip_runtime.h>

typedef __attribute__((ext_vector_type(2))) float v2f;
typedef __attribute__((ext_vector_type(8))) float v8f;

#define Hh 128
#define Ww 128
#define HW (Hh * Ww)
#define Bb 16
#define Cc 16
#define PATCHES 4
#define KDIM 256        // N_CTRL * M_CTRL
#define S_PER_PATCH 100
#define S_PAD 112       // 7 tiles of 16, zero-padded rows 100..111
#define S_TOT 400
#define L_TOT 1024      // PATCH_NUM * N_CTRL * M_CTRL
#define BLOCK 256
#define BST 260         // padded LDS row stride (260 mod 64 banks = 4 -> conflict-free)

// ---------------- kernel 0: zero the workspace accumulators ----------------
__global__ void tl_init_ws(float* ws) {
    if (threadIdx.x < 3) ws[threadIdx.x] = 0.0f;
}

// ---------------- kernel 1: loss1 numerator + count -----------------------
// loss1_num = sum_{b,c,h,w} | (input[b,hw]!=0 ? output : 0) - target |
// cnt       = sum_{b,h,w} input[b,hw]
__global__ void tl_loss1(const float* __restrict__ inp,
                         const float* __restrict__ tgt,
                         const float* __restrict__ out,
                         float* __restrict__ ws) {
    __shared__ float red[BLOCK];
    const int tid = threadIdx.x;
    long long gid = (long long)blockIdx.x * BLOCK + tid;
    const long long stride = (long long)gridDim.x * BLOCK;
    const long long NSP = (long long)Bb * HW;

    float acc = 0.0f, accCnt = 0.0f;
    for (long long i = gid; i < NSP; i += stride) {
        long long b = i / HW;
        long long hw = i - b * HW;
        float m = inp[i];                 // input is (B,1,H,W): flat == i
        accCnt += m;
        bool nz = (m != 0.0f);
        const float* tb = tgt + b * (long long)Cc * HW + hw;
        const float* ob = out + b * (long long)Cc * HW + hw;
#pragma unroll
        for (int c = 0; c < Cc; ++c) {
            float t = tb[(long long)c * HW];
            float o = ob[(long long)c * HW];
            float d = nz ? (o - t) : (-t);
            acc += fabsf(d);
        }
    }

    red[tid] = acc;
    __syncthreads();
    for (int s = BLOCK / 2; s > 0; s >>= 1) {
        if (tid < s) red[tid] += red[tid + s];
        __syncthreads();
    }
    if (tid == 0) atomicAdd(&ws[0], red[0]);
    __syncthreads();

    red[tid] = accCnt;
    __syncthreads();
    for (int s = BLOCK / 2; s > 0; s >>= 1) {
        if (tid < s) red[tid] += red[tid + s];
        __syncthreads();
    }
    if (tid == 0) atomicAdd(&ws[1], red[0]);
}

// ---------------- kernel 2: loss2 numerator via f32 WMMA -------------------
// One block per (batch b, patch p). Stage diff G[c][k] (gathered at mapRecord
// sites) and the patch's basis tile (zero-padded to 112 samples, columns
// permuted [c0,c2,c1,c3] per 4-group so each lane's B pair is contiguous)
// into padded LDS. Waves 0..6 each own a 16-sample N tile and run 64 chained
// V_WMMA_F32_16X16X4_F32 steps; inner loop = 2x ds_load_b64 + 1x wmma,
// branchless, conflict-free.
__global__ void tl_loss2_wmma(const int* __restrict__ mapRecord,
                              const float* __restrict__ tgt,
                              const float* __restrict__ out,
                              const float* __restrict__ basis,
                              float* __restrict__ ws) {
    __shared__ float G[Cc * BST];        // ~16.6 KB
    __shared__ float BS[S_PAD * BST];    // ~116.5 KB
    __shared__ float red[BLOCK];

    const int tid = threadIdx.x;
    const int b = blockIdx.x / PATCHES;
    const int p = blockIdx.x % PATCHES;

    // ---- stage basis tile (float4 loads, permuted+padded stores) ----
    for (int j = tid; j < S_PAD * (KDIM / 4); j += BLOCK) {
        const int row = j >> 6;            // j / (KDIM/4)
        const int c4  = (j & 63) << 2;     // column group base
        float4 v = make_float4(0.f, 0.f, 0.f, 0.f);
        if (row < S_PER_PATCH) {
            const float* src = basis + ((long long)(p * S_PER_PATCH + row) * KDIM + c4);
            v = *(const float4*)src;
        }
        float4 perm = make_float4(v.x, v.z, v.y, v.w);   // [c0,c2,c1,c3]
        *(float4*)&BS[row * BST + c4] = perm;
    }

    // ---- gather phase: thread tid owns k = tid (one control point) ----
    {
        const int k = tid;                       // 0..255
        const int l = p * KDIM + k;              // index into L dimension
        const int* mr = mapRecord + ((long long)b * L_TOT + l) * 2;
        const int flat = mr[0] * Ww + mr[1];
        const long long base = (long long)b * Cc * HW + flat;
#pragma unroll
        for (int c = 0; c < Cc; ++c) {
            long long idx = base + (long long)c * HW;
            G[c * BST + k] = tgt[idx] - out[idx];
        }
    }
    __syncthreads();

    const int wave = tid >> 5;
    const int lane = tid & 31;
    float acc = 0.0f;

    if (wave < 7) {                       // 7 tiles of 16 cover S_PER_PATCH=100
        const int n  = lane & 15;         // column within tile (sample)
        const int hi = lane >> 4;         // selects K sub-pair per ISA layout
        const int m  = lane & 15;         // A-matrix row (channel)
        const int sLocal = wave * 16 + n;
        const float* gRow = &G[m * BST];
        const float* bRow = &BS[sLocal * BST];

        v8f d = {};                       // fp32 accumulator (C/D, 8 VGPRs)
        for (int k0 = 0; k0 < KDIM; k0 += 4) {
            // A 16x4 f32: lane holds A[m][k0+2*hi+{0,1}] -> one b64
            v2f a = *(const v2f*)(gRow + k0 + 2 * hi);
            // B 4x16 f32: permuted layout puts (B[k0+hi], B[k0+2+hi]) at
            // contiguous offset 2*hi -> one b64
            v2f bv = *(const v2f*)(bRow + k0 + 2 * hi);
            d = __builtin_amdgcn_wmma_f32_16x16x4_f32(
                    /*neg_a=*/false, a, /*neg_b=*/false, bv,
                    /*c_mod=*/(short)0, d, /*reuse_a=*/false, /*reuse_b=*/false);
        }
        // lane's D entries: rows v + 8*hi, column n == sLocal (if valid)
        if (sLocal < S_PER_PATCH) {
#pragma unroll
            for (int v = 0; v < 8; ++v) acc += fabsf(d[v]);
        }
    }

    red[tid] = acc;
    __syncthreads();
    for (int s = BLOCK / 2; s > 0; s >>= 1) {
        if (tid < s) red[tid] += red[tid + s];
        __syncthreads();
    }
    if (tid == 0) atomicAdd(&ws[2], red[0]);
}

// ---------------- kernel 3: finalize the three scalars ---------------------
__global__ void tl_finalize(const float* __restrict__ ws, float* __restrict__ o) {
    float loss1 = ws[0] / (ws[1] * (float)Cc);
    float loss2 = ws[2] / (float)((long long)Bb * Cc * S_TOT);
    o[0] = loss1 + loss2;   // W1 = W2 = 1
    o[1] = loss1;
    o[2] = loss2;
}

extern "C" void kernel_launch(void* const* d_in, const int* in_sizes, int n_in,
                              void* d_out, int out_size, void* d_ws, size_t ws_size,
                              hipStream_t stream) {
    const float* inp       = (const float*)d_in[0]; // (16,1,128,128)
    const int*   mapRecord = (const int*)  d_in[1]; // (16,1024,2)
    const float* tgt       = (const float*)d_in[2]; // (16,16,128,128)
    const float* out       = (const float*)d_in[3]; // (16,16,128,128)
    const float* basis     = (const float*)d_in[4]; // (400,16,16)
    // d_in[5] = patchIndex: repeat(arange(4),100) by construction; tiling by
    // patch (100 samples each) encodes it directly.
    (void)in_sizes; (void)n_in; (void)out_size; (void)ws_size;

    float* ws = (float*)d_ws;   // ws[0]=loss1 num, ws[1]=sum(input), ws[2]=loss2 num
    float* o  = (float*)d_out;  // 3 floats: loss, loss1, loss2

    tl_init_ws<<<1, 32, 0, stream>>>(ws);
    tl_loss1<<<1024, BLOCK, 0, stream>>>(inp, tgt, out, ws);
    tl_loss2_wmma<<<Bb * PATCHES, BLOCK, 0, stream>>>(mapRecord, tgt, out, basis, ws);
    tl_finalize<<<1, 1, 0, stream>>>(ws, o);
}